// ProgressiveFocusedSparseAttention_34806414966815
// MI455X (gfx1250) — compile-verified
//
#include <hip/hip_runtime.h>
#include <math.h>

// Problem constants (from reference)
#define WB     2
#define WN     4096
#define WDIM   512
#define WH     8
#define WGS    64
#define WNG    64      // N / GS
#define WD     64      // head dim
#define WM     64      // prototypes
#define WNL    16      // labels
#define TWO_GS 128

typedef __attribute__((ext_vector_type(2))) float v2f;
typedef __attribute__((ext_vector_type(8))) float v8f;
typedef __attribute__((ext_vector_type(4))) int   pfsa_v4i;

// ---------------------------------------------------------------------------
// CDNA5 async global->LDS path (ASYNCcnt-tracked). Fallback: sync float4 copy.
// Builtin signature (from clang diagnostic): param0 = int4 addrspace(1)*,
// i.e. (global src, lds dst, imm offset, imm cpol).
// ---------------------------------------------------------------------------
#if defined(__AMDGCN__)
  #if __has_builtin(__builtin_amdgcn_global_load_async_to_lds_b128)
    #define PFSA_ASYNC 1
    #warning "PFSA: using global_load_async_to_lds_b128 staging"
  #else
    #warning "PFSA: async-to-LDS builtin unavailable, using sync float4 staging"
  #endif
#endif

#ifdef PFSA_ASYNC
#define PFSA_CP128(dst, src)                                                   \
  __builtin_amdgcn_global_load_async_to_lds_b128(                              \
      (__attribute__((address_space(1))) pfsa_v4i*)(uintptr_t)(src),           \
      (__attribute__((address_space(3))) pfsa_v4i*)(uintptr_t)(dst), 0, 0)
static __device__ __forceinline__ void pfsa_async_wait0() {
  #if __has_builtin(__builtin_amdgcn_s_wait_asynccnt)
  __builtin_amdgcn_s_wait_asynccnt(0);
  #else
  asm volatile("s_wait_asynccnt 0x0" ::: "memory");
  #endif
}
#else
#define PFSA_CP128(dst, src) (*(float4*)(dst) = *(const float4*)(src))
static __device__ __forceinline__ void pfsa_async_wait0() {}
#endif

static __device__ __forceinline__ float waveRedMax(float v) {
#pragma unroll
  for (int off = 16; off > 0; off >>= 1) v = fmaxf(v, __shfl_xor(v, off, 32));
  return v;
}
static __device__ __forceinline__ float waveRedSum(float v) {
#pragma unroll
  for (int off = 16; off > 0; off >>= 1) v += __shfl_xor(v, off, 32);
  return v;
}

// ---------------------------------------------------------------------------
// Generic fp32 WMMA GEMM: C[M x N] = A[M x K] * B[K x N]
// Block = 256 threads (8 wave32s) -> 32x64 C tile (2x4 waves of 16x16).
// K chunks of 32 staged via async-to-LDS with double buffering, consumed by
// V_WMMA_F32_16X16X4_F32.
// ---------------------------------------------------------------------------
__global__ __launch_bounds__(256) void pfsa_gemm_f32(const float* __restrict__ A,
                                                     const float* __restrict__ Bm,
                                                     float* __restrict__ C,
                                                     int M, int Nn, int K) {
  __shared__ float As[2][32][36];   // row stride 144B (16B aligned, odd banks)
  __shared__ float Bs[2][32][68];   // row stride 272B
  const int tid  = threadIdx.x;
  const int wave = tid >> 5;
  const int lane = tid & 31;
  const int lx   = lane & 15;
  const int hi   = lane >> 4;
  const int wm   = wave >> 2;   // 0..1
  const int wn   = wave & 3;    // 0..3
  const int rowBase = blockIdx.y * 32;
  const int colBase = blockIdx.x * 64;

  // per-thread staging indices
  const int ar = tid >> 3, ac = (tid & 7) * 4;        // A tile 32x32: 1 f4
  const int br = tid >> 4, bc = (tid & 15) * 4;       // B tile 32x64: 2 f4
  const int br2 = br + 16;

  auto stage = [&](int bufi, int kk) {
    PFSA_CP128(&As[bufi][ar][ac], &A[(size_t)(rowBase + ar) * K + kk + ac]);
    PFSA_CP128(&Bs[bufi][br][bc], &Bm[(size_t)(kk + br) * Nn + colBase + bc]);
    PFSA_CP128(&Bs[bufi][br2][bc], &Bm[(size_t)(kk + br2) * Nn + colBase + bc]);
  };

  v8f acc = {0.f, 0.f, 0.f, 0.f, 0.f, 0.f, 0.f, 0.f};

  stage(0, 0);
  pfsa_async_wait0();
  __syncthreads();

  int buf = 0;
  for (int kk = 0; kk < K; kk += 32) {
    if (kk + 32 < K) stage(buf ^ 1, kk + 32);   // overlap next chunk (ASYNCcnt)
#pragma unroll
    for (int k4 = 0; k4 < 32; k4 += 4) {
      v2f a, b;
      a.x = As[buf][wm * 16 + lx][k4 + 2 * hi];
      a.y = As[buf][wm * 16 + lx][k4 + 2 * hi + 1];
      b.x = Bs[buf][k4 + 2 * hi][wn * 16 + lx];
      b.y = Bs[buf][k4 + 2 * hi + 1][wn * 16 + lx];
      acc = __builtin_amdgcn_wmma_f32_16x16x4_f32(false, a, false, b, (short)0, acc,
                                                  false, false);
    }
    pfsa_async_wait0();
    __syncthreads();
    buf ^= 1;
  }
#pragma unroll
  for (int r = 0; r < 8; ++r) {
    int row = rowBase + wm * 16 + r + 8 * hi;
    int col = colBase + wn * 16 + lx;
    C[(size_t)row * Nn + col] = acc[r];
  }
}

// ---------------------------------------------------------------------------
// Fused windowed attention. One workgroup (512 threads = 16 wave32s) per
// (b, g, h) window. Entire window state lives in LDS (~118 KB < 320 KB/WGP):
//   q 64x64, k/v 128x64, attn 64x128; kg/vg/gl overlap dead k/attn regions.
// ---------------------------------------------------------------------------
__global__ __launch_bounds__(512) void pfsa_attn_kernel(
    const float* __restrict__ qg, const float* __restrict__ kgl,
    const float* __restrict__ vgl, const int* __restrict__ labels,
    const float* __restrict__ scores, const float* __restrict__ prev,
    const float* __restrict__ kproj, const float* __restrict__ vproj,
    const float* __restrict__ beta_p, float* __restrict__ out_pre,
    float* __restrict__ attn_out) {
  __shared__ float smem[30208];
  float(*q_s)[68]   = (float(*)[68])(smem);                  // 64 x 68   @ 0
  float(*k_s)[68]   = (float(*)[68])(smem + 4352);           // 128 x 68
  float(*v_s)[68]   = (float(*)[68])(smem + 13056);          // 128 x 68
  float(*at_s)[132] = (float(*)[132])(smem + 21760);         // 64 x 132
  // overlapped regions (dead-data reuse):
  float(*kg_s)[68] = (float(*)[68])(smem + 4352);            // over k_s lo
  float(*vg_s)[68] = (float(*)[68])(smem + 8704);            // over k_s hi
  float(*gl_s)[68] = (float(*)[68])(smem + 21760);           // over at_s
  __shared__ int   ql_s[64];
  __shared__ int   kl_s[128];
  __shared__ float qs_s[64];
  __shared__ int   keep_sh;
  __shared__ float beta_sh;

  const int tid  = threadIdx.x;
  const int wave = tid >> 5;
  const int lane = tid & 31;
  const int lx   = lane & 15;
  const int hi   = lane >> 4;

  int wg = blockIdx.x;
  const int h = wg % WH;  wg /= WH;
  const int g = wg % WNG; wg /= WNG;
  const int b = wg;

  const size_t headBase = (size_t)b * WN * WDIM + (size_t)h * WD;

  // ---- stage window into LDS (async b128, 16B-aligned both sides) ----
#pragma unroll
  for (int t = tid; t < WGS * WD / 4; t += 512) {       // q: 1024 f4
    int i = t >> 4, dd = (t & 15) * 4;
    PFSA_CP128(&q_s[i][dd], &qg[headBase + (size_t)(g * WGS + i) * WDIM + dd]);
  }
#pragma unroll
  for (int t = tid; t < TWO_GS * WD / 4; t += 512) {    // k,v: 2048 f4 each
    int j = t >> 4, dd = (t & 15) * 4;
    int tok = (j < WGS) ? (g * WGS + j)
                        : ((g < WNG - 1) ? (g * WGS + j) : (WN - 1 - (j - WGS)));
    PFSA_CP128(&k_s[j][dd], &kgl[headBase + (size_t)tok * WDIM + dd]);
    PFSA_CP128(&v_s[j][dd], &vgl[headBase + (size_t)tok * WDIM + dd]);
  }
  for (int t = tid; t < TWO_GS; t += 512) {
    int tok = (t < WGS) ? (g * WGS + t)
                        : ((g < WNG - 1) ? (g * WGS + t) : (WN - 1 - (t - WGS)));
    kl_s[t] = labels[b * WN + tok];
  }
  for (int t = tid; t < WGS; t += 512) {
    ql_s[t] = labels[b * WN + g * WGS + t];
    qs_s[t] = scores[b * WN + g * WGS + t];
  }
  if (tid == 0) beta_sh = 1.0f / (1.0f + __expf(-beta_p[0]));
  pfsa_async_wait0();
  __syncthreads();

  const float scale = 0.125f;  // d^-0.5, d=64

  // ---- logits = qw . kw^T, masked; 32 tiles of 16x16, 2 per wave ----
#pragma unroll
  for (int s = 0; s < 2; ++s) {
    int tile = wave * 2 + s;
    int tm = tile >> 3, tk = tile & 7;
    v8f acc = {0.f, 0.f, 0.f, 0.f, 0.f, 0.f, 0.f, 0.f};
#pragma unroll
    for (int k4 = 0; k4 < WD; k4 += 4) {
      v2f a, bb;
      a.x  = q_s[tm * 16 + lx][k4 + 2 * hi];
      a.y  = q_s[tm * 16 + lx][k4 + 2 * hi + 1];
      bb.x = k_s[tk * 16 + lx][k4 + 2 * hi];      // B[d][key] = kw[key][d]
      bb.y = k_s[tk * 16 + lx][k4 + 2 * hi + 1];
      acc = __builtin_amdgcn_wmma_f32_16x16x4_f32(false, a, false, bb, (short)0, acc,
                                                  false, false);
    }
#pragma unroll
    for (int r = 0; r < 8; ++r) {
      int row = tm * 16 + r + 8 * hi;
      int col = tk * 16 + lx;
      float lv = acc[r] * scale;
      at_s[row][col] = (ql_s[row] == kl_s[col]) ? lv : -10000.0f;
    }
  }
  __syncthreads();

  // ---- scalar per-window focus ratio (overlaps softmax on other lanes) ----
  if (tid == 0) {
    int cnt[WNL];
#pragma unroll
    for (int i = 0; i < WNL; ++i) cnt[i] = 0;
    for (int i = 0; i < WGS; ++i) cnt[ql_s[i] & (WNL - 1)]++;
    int bestc = -1;
    for (int l = 0; l < WNL; ++l)
      if (cnt[l] > bestc) bestc = cnt[l];
    float purity = (float)bestc * (1.0f / WGS);
    float s1 = 0.f, s2 = 0.f;
    for (int i = 0; i < WGS; ++i) { float sc = qs_s[i]; s1 += sc; s2 += sc * sc; }
    float mean = s1 * (1.0f / WGS);
    float var  = s2 * (1.0f / WGS) - mean * mean;
    float focus = fminf(fmaxf(0.5f + 0.25f * purity - 0.25f * var, 0.25f), 0.75f);
    int kp = (int)ceilf(focus * (float)TWO_GS);
    keep_sh = min(max(kp, 1), TWO_GS);
  }

  // ---- row softmax, cluster renorm, beta blend, renorm ----
  // Lane owns 4 contiguous columns (lane*4..lane*4+3) => b128 LDS/global ops.
  const float beta = beta_sh;
#pragma unroll 1
  for (int rr = 0; rr < 4; ++rr) {
    int row = wave * 4 + rr;
    const int c0 = lane * 4;
    float vals[4];
    bool  cl[4];
    int   qlab = ql_s[row];
    float4 vv = *(const float4*)&at_s[row][c0];
    vals[0] = vv.x; vals[1] = vv.y; vals[2] = vv.z; vals[3] = vv.w;
    float mx = fmaxf(fmaxf(vals[0], vals[1]), fmaxf(vals[2], vals[3]));
    mx = waveRedMax(mx);
    float se = 0.f;
#pragma unroll
    for (int j = 0; j < 4; ++j) { vals[j] = __expf(vals[j] - mx); se += vals[j]; }
    se = waveRedSum(se);
    float inv = 1.0f / se, sm = 0.f;
#pragma unroll
    for (int j = 0; j < 4; ++j) {
      cl[j]   = (qlab == kl_s[c0 + j]);
      vals[j] = cl[j] ? vals[j] * inv : 0.f;
      sm += vals[j];
    }
    sm = waveRedSum(sm);
    float inv2 = 1.0f / (sm + 1e-9f);
    const float* prow =
        prev + (((size_t)(b * WNG + g) * WH + h) * WGS + row) * TWO_GS;
    float4 pv = *(const float4*)&prow[c0];
    float pr[4] = {pv.x, pv.y, pv.z, pv.w};
    float sb = 0.f;
#pragma unroll
    for (int j = 0; j < 4; ++j) {
      float a = beta * pr[j] + (1.f - beta) * (vals[j] * inv2);
      a = cl[j] ? a : 0.f;
      vals[j] = a;
      sb += a;
    }
    sb = waveRedSum(sb);
    float inv3 = 1.0f / (sb + 1e-9f);
    float4 ov = make_float4(vals[0] * inv3, vals[1] * inv3, vals[2] * inv3,
                            vals[3] * inv3);
    *(float4*)&at_s[row][c0] = ov;
  }
  __syncthreads();

  // ---- dynamic top-k via exact stable rank; renorm; write attn_sparse ----
  const int   keep = keep_sh;
  float* attn_base =
      attn_out + (((size_t)(b * WNG + g) * WH + h) * WGS) * TWO_GS;
#pragma unroll 1
  for (int rr = 0; rr < 4; ++rr) {
    int row = wave * 4 + rr;
    const int c0 = lane * 4;
    float vals[4];
    int   ranks[4];
    float4 vv = *(const float4*)&at_s[row][c0];
    vals[0] = vv.x; vals[1] = vv.y; vals[2] = vv.z; vals[3] = vv.w;
#pragma unroll
    for (int j = 0; j < 4; ++j) ranks[j] = 0;
    for (int e = 0; e < TWO_GS; ++e) {
      float fe = at_s[row][e];
#pragma unroll
      for (int j = 0; j < 4; ++j)
        ranks[j] += (fe > vals[j] || (fe == vals[j] && e < c0 + j)) ? 1 : 0;
    }
    float ssum = 0.f;
#pragma unroll
    for (int j = 0; j < 4; ++j) {
      vals[j] = (ranks[j] < keep) ? vals[j] : 0.f;
      ssum += vals[j];
    }
    ssum = waveRedSum(ssum);
    float inv = 1.0f / (ssum + 1e-9f);
    float4 ov = make_float4(vals[0] * inv, vals[1] * inv, vals[2] * inv,
                            vals[3] * inv);
    *(float4*)&at_s[row][c0] = ov;                    // same wave owns the row
    *(float4*)&attn_base[row * TWO_GS + c0] = ov;     // global b128 store
  }
  __syncthreads();

  // ---- out1 = attn(64x128) . vw(128x64); one 16x16 tile per wave ----
  const int tm = wave >> 2, tn = wave & 3;
  v8f oacc = {0.f, 0.f, 0.f, 0.f, 0.f, 0.f, 0.f, 0.f};
#pragma unroll
  for (int k4 = 0; k4 < TWO_GS; k4 += 4) {
    v2f a, bb;
    a.x  = at_s[tm * 16 + lx][k4 + 2 * hi];
    a.y  = at_s[tm * 16 + lx][k4 + 2 * hi + 1];
    bb.x = v_s[k4 + 2 * hi][tn * 16 + lx];
    bb.y = v_s[k4 + 2 * hi + 1][tn * 16 + lx];
    oacc = __builtin_amdgcn_wmma_f32_16x16x4_f32(false, a, false, bb, (short)0, oacc,
                                                 false, false);
  }
  __syncthreads();

  // ---- stage prototype K/V heads into (dead) k_s region ----
#pragma unroll
  for (int t = tid; t < WM * WD / 4; t += 512) {
    int mrow = t >> 4, dd = (t & 15) * 4;
    PFSA_CP128(&kg_s[mrow][dd], &kproj[(size_t)(b * WM + mrow) * WDIM + h * WD + dd]);
    PFSA_CP128(&vg_s[mrow][dd], &vproj[(size_t)(b * WM + mrow) * WDIM + h * WD + dd]);
  }
  pfsa_async_wait0();
  __syncthreads();

  // ---- gl = qw . kg^T (scaled) into gl_s (over dead at_s) ----
  {
    v8f gacc = {0.f, 0.f, 0.f, 0.f, 0.f, 0.f, 0.f, 0.f};
#pragma unroll
    for (int k4 = 0; k4 < WD; k4 += 4) {
      v2f a, bb;
      a.x  = q_s[tm * 16 + lx][k4 + 2 * hi];
      a.y  = q_s[tm * 16 + lx][k4 + 2 * hi + 1];
      bb.x = kg_s[tn * 16 + lx][k4 + 2 * hi];
      bb.y = kg_s[tn * 16 + lx][k4 + 2 * hi + 1];
      gacc = __builtin_amdgcn_wmma_f32_16x16x4_f32(false, a, false, bb, (short)0, gacc,
                                                   false, false);
    }
#pragma unroll
    for (int r = 0; r < 8; ++r)
      gl_s[tm * 16 + r + 8 * hi][tn * 16 + lx] = gacc[r] * scale;
  }
  __syncthreads();

  // ---- softmax over prototypes (64 wide, 2 elems/lane) ----
#pragma unroll 1
  for (int rr = 0; rr < 4; ++rr) {
    int row = wave * 4 + rr;
    float v0 = gl_s[row][lane], v1 = gl_s[row][lane + 32];
    float mx = waveRedMax(fmaxf(v0, v1));
    v0 = __expf(v0 - mx);
    v1 = __expf(v1 - mx);
    float s = waveRedSum(v0 + v1);
    float inv = 1.0f / s;
    gl_s[row][lane] = v0 * inv;
    gl_s[row][lane + 32] = v1 * inv;
  }
  __syncthreads();

  // ---- out += softmax(gl) . vg ----
#pragma unroll
  for (int k4 = 0; k4 < WD; k4 += 4) {
    v2f a, bb;
    a.x  = gl_s[tm * 16 + lx][k4 + 2 * hi];
    a.y  = gl_s[tm * 16 + lx][k4 + 2 * hi + 1];
    bb.x = vg_s[k4 + 2 * hi][tn * 16 + lx];
    bb.y = vg_s[k4 + 2 * hi + 1][tn * 16 + lx];
    oacc = __builtin_amdgcn_wmma_f32_16x16x4_f32(false, a, false, bb, (short)0, oacc,
                                                 false, false);
  }

  // ---- write pre-projection output [B, N, DIM] ----
#pragma unroll
  for (int r = 0; r < 8; ++r) {
    int row = tm * 16 + r + 8 * hi;
    int col = tn * 16 + lx;
    out_pre[(size_t)b * WN * WDIM + (size_t)(g * WGS + row) * WDIM + h * WD + col] =
        oacc[r];
  }
}

// ---------------------------------------------------------------------------
extern "C" void kernel_launch(void* const* d_in, const int* in_sizes, int n_in,
                              void* d_out, int out_size, void* d_ws, size_t ws_size,
                              hipStream_t stream) {
  (void)in_sizes; (void)n_in; (void)out_size; (void)ws_size;
  const float* x      = (const float*)d_in[0];
  const int*   labels = (const int*)d_in[1];
  const float* scores = (const float*)d_in[2];
  const float* prev   = (const float*)d_in[3];
  const float* protos = (const float*)d_in[4];
  const float* Wq     = (const float*)d_in[5];
  const float* Wk     = (const float*)d_in[6];
  const float* Wv     = (const float*)d_in[7];
  const float* Wproj  = (const float*)d_in[8];
  const float* beta   = (const float*)d_in[9];

  float* out      = (float*)d_out;                       // [B,N,DIM]
  float* attn_out = out + (size_t)WB * WN * WDIM;        // [B,ng,H,GS,2GS]

  float*  ws  = (float*)d_ws;
  size_t  big = (size_t)WB * WN * WDIM;                  // 4,194,304 floats
  float* qb = ws;
  float* kb = qb + big;
  float* vb = kb + big;
  float* ob = vb + big;
  float* kp = ob + big;
  float* vp = kp + (size_t)WB * WM * WDIM;

  dim3 blk(256);
  dim3 gBig(WDIM / 64, (WB * WN) / 32);   // (8, 256)
  dim3 gSm(WDIM / 64, (WB * WM) / 32);    // (8, 4)

  pfsa_gemm_f32<<<gBig, blk, 0, stream>>>(x, Wq, qb, WB * WN, WDIM, WDIM);
  pfsa_gemm_f32<<<gBig, blk, 0, stream>>>(x, Wk, kb, WB * WN, WDIM, WDIM);
  pfsa_gemm_f32<<<gBig, blk, 0, stream>>>(x, Wv, vb, WB * WN, WDIM, WDIM);
  pfsa_gemm_f32<<<gSm,  blk, 0, stream>>>(protos, Wk, kp, WB * WM, WDIM, WDIM);
  pfsa_gemm_f32<<<gSm,  blk, 0, stream>>>(protos, Wv, vp, WB * WM, WDIM, WDIM);

  pfsa_attn_kernel<<<WB * WNG * WH, 512, 0, stream>>>(
      qb, kb, vb, labels, scores, prev, kp, vp, beta, ob, attn_out);

  pfsa_gemm_f32<<<gBig, blk, 0, stream>>>(ob, Wproj, out, WB * WN, WDIM, WDIM);
}